// GNNLayer_67577015435433
// MI455X (gfx1250) — compile-verified
//
#include <hip/hip_runtime.h>
#include <math.h>

// ---------------------------------------------------------------------------
// GatedGCN layer, fused, fp32 end-to-end using CDNA5 V_WMMA_F32_16X16X4_F32.
// B=8, N=256, D=128, F=512.  Bandwidth-bound: ~1.25 GB of e-traffic ~= 55us
// at 23.3 TB/s; 17.2 GFLOP of GEMM hides under the stream on the WMMA pipe.
// d_out e-region doubles as the e_new temp (BN needs global stats -> 2 pass).
// Workspace use: ~6.3 MB.
// ---------------------------------------------------------------------------

#define D   128
#define F   512
#define NB  8
#define NN  256
#define MH  (NB * NN)        // 2048 rows of h
#define ME  (NB * NN * NN)   // 524288 rows of e
#define EPSQ 1e-5f

typedef __attribute__((ext_vector_type(2))) float v2f;
typedef __attribute__((ext_vector_type(8))) float v8f;

// D(16x16,f32) = A(16x4,f32) x B(4x16,f32) + C
// A frag: lane l holds M=l%16, K = v + 2*(l/16)   (v = vgpr 0/1)
// B frag: lane l holds N=l%16, K = v + 2*(l/16)
// C/D:    lane l, vgpr r holds M = r + 8*(l/16), N = l%16
__device__ __forceinline__ v8f wmma_f32(v2f a, v2f b, v8f c) {
  return __builtin_amdgcn_wmma_f32_16x16x4_f32(false, a, false, b, (short)0, c,
                                               false, false);
}

__device__ __forceinline__ float sigmoidf_(float x) {
  return 1.f / (1.f + __expf(-x));
}

// ---------------------------------------------------------------- k0: zero ws
__global__ void k0_zero(float* __restrict__ p, int n) {
  int i = blockIdx.x * blockDim.x + threadIdx.x;
  if (i < n) p[i] = 0.f;
}

// ------------------------------------------------- k1: Uh,Vh,Ah,Bh = h@W + b
__global__ __launch_bounds__(256) void k1_proj(
    const float* __restrict__ h,
    const float* __restrict__ Uw, const float* __restrict__ Ub,
    const float* __restrict__ Vw, const float* __restrict__ Vb,
    const float* __restrict__ Aw, const float* __restrict__ Ab,
    const float* __restrict__ Bw, const float* __restrict__ Bb,
    float* __restrict__ out4)  // 4 contiguous [MH, D] buffers
{
  const int tid = threadIdx.x, wave = tid >> 5, lane = tid & 31;
  const int half = lane >> 4, mr = lane & 15;
  const int tile = blockIdx.x * 8 + wave;   // 4096 tiles total
  const int g  = tile >> 10;                // which weight matrix
  const int t  = tile & 1023;
  const int m0 = (t >> 3) * 16;
  const int n0 = (t & 7) * 16;

  const float* W; const float* bias;
  if      (g == 0) { W = Uw; bias = Ub; }
  else if (g == 1) { W = Vw; bias = Vb; }
  else if (g == 2) { W = Aw; bias = Ab; }
  else             { W = Bw; bias = Bb; }

  v8f acc = {};
  for (int k = 0; k < D; k += 4) {
    const int kk = k + 2 * half;
    v2f a = *(const v2f*)(h + (m0 + mr) * D + kk);
    v2f b; b.x = W[kk * D + n0 + mr]; b.y = W[(kk + 1) * D + n0 + mr];
    acc = wmma_f32(a, b, acc);
  }
  const float bb = bias[n0 + mr];
  float* op = out4 + (size_t)g * MH * D;
#pragma unroll
  for (int r = 0; r < 8; ++r)
    op[(m0 + r + 8 * half) * D + n0 + mr] = acc[r] + bb;
}

// ---- k2: e_new = e@Cw + Cb + Ah[b,j] + Bh[b,i]; gate-sum into h_agg; stats
__global__ __launch_bounds__(512) void k2_edge(
    const float* __restrict__ e,  const float* __restrict__ Cw,
    const float* __restrict__ Cb, const float* __restrict__ Ah,
    const float* __restrict__ Bh, const float* __restrict__ Vh,
    float* __restrict__ e_tmp,    // = d_out e-region (temp e_new storage)
    float* __restrict__ h_agg, float* __restrict__ sum_e,
    float* __restrict__ sumsq_e)
{
  // Cw staged transposed, half of K at a time: cwT[n][kk], pitch 68 dwords
  // (68 % 64 == 4 -> float2 B-frag reads are LDS bank-conflict free).
  __shared__ float cwT[D * 68];
  __shared__ float ls[D], lss[D];

  const int tid = threadIdx.x, wave = tid >> 5, lane = tid & 31;
  const int half = lane >> 4, mr = lane & 15;
  const int blk = blockIdx.x;           // 2048 = 8 b * 16 itile * 16 jtile
  const int b  = blk >> 8;
  const int it = (blk >> 4) & 15;
  const int jt = blk & 15;
  const int i  = it * 16 + wave;        // one i per wave
  const int j0 = jt * 16;

  if (tid < D) { ls[tid] = 0.f; lss[tid] = 0.f; }

  v8f acc[8] = {};                      // 8 n-tiles -> full 128 output cols
  // A rows: e[b,i,j0..j0+15,:], contiguous with stride D
  const float* erow = e + (((size_t)(b * NN + i)) * NN + j0) * D;

  for (int kh = 0; kh < 2; ++kh) {
    __syncthreads();
    for (int idx = tid; idx < 64 * D; idx += 512) {
      const int kk = idx >> 7, n = idx & 127;
      cwT[n * 68 + kk] = Cw[(kh * 64 + kk) * D + n];
    }
    __syncthreads();
    for (int k = 0; k < 64; k += 4) {
      const int kk = k + 2 * half;
      v2f a = *(const v2f*)(erow + mr * D + kh * 64 + kk);
#pragma unroll
      for (int t = 0; t < 8; ++t) {
        v2f bf = *(const v2f*)(&cwT[(t * 16 + mr) * 68 + kk]);
        acc[t] = wmma_f32(a, bf, acc[t]);
      }
    }
  }

  // Epilogue: bias + broadcast adds, store e_new, gate*Vh sum, BN stats.
  const float* BhRow = Bh + (b * NN + i) * D;
  float* etile = e_tmp + (((size_t)(b * NN + i)) * NN + j0) * D;
#pragma unroll
  for (int t = 0; t < 8; ++t) {
    const int n = t * 16 + mr;
    const float bias = Cb[n] + BhRow[n];
    float cs = 0.f, css = 0.f, hacc = 0.f;
#pragma unroll
    for (int r = 0; r < 8; ++r) {
      const int j = j0 + r + 8 * half;
      float v = acc[t][r] + bias + Ah[(b * NN + j) * D + n];
      etile[(r + 8 * half) * D + n] = v;
      cs += v; css += v * v;
      hacc += sigmoidf_(v) * Vh[(b * NN + j) * D + n];
    }
    // lanes l and l+16 hold the same (i,n) over complementary j halves
    hacc += __shfl_xor(hacc, 16);
    cs   += __shfl_xor(cs, 16);
    css  += __shfl_xor(css, 16);
    if (half == 0) {
      atomicAdd(&h_agg[(b * NN + i) * D + n], hacc);
      atomicAdd(&ls[n], cs);
      atomicAdd(&lss[n], css);
    }
  }
  __syncthreads();
  if (tid < D) {
    atomicAdd(&sum_e[tid], ls[tid]);
    atomicAdd(&sumsq_e[tid], lss[tid]);
  }
}

// -------------------------- k3a: h_pre = Uh + h_agg ; BN stats for h columns
__global__ __launch_bounds__(256) void k3a_hstats(
    const float* __restrict__ Uh, const float* __restrict__ h_agg,
    float* __restrict__ h_pre, float* __restrict__ sum_h,
    float* __restrict__ sumsq_h)
{
  __shared__ float ls[256], lss[256];
  const int tid = threadIdx.x;
  const int col = tid & 127, rg = tid >> 7;
  const int r0 = blockIdx.x * 32;
  float s = 0.f, ss = 0.f;
  for (int r = rg; r < 32; r += 2) {
    const int idx = (r0 + r) * D + col;
    const float v = Uh[idx] + h_agg[idx];
    h_pre[idx] = v;
    s += v; ss += v * v;
  }
  ls[tid] = s; lss[tid] = ss;
  __syncthreads();
  if (tid < 128) {
    atomicAdd(&sum_h[col], ls[tid] + ls[tid + 128]);
    atomicAdd(&sumsq_h[col], lss[tid] + lss[tid + 128]);
  }
}

// ------- k3b: fused BN(h) -> W1 -> GELU -> W2 -> +h_in, one 16-row tile / WG
__global__ __launch_bounds__(256) void k3b_ffn(
    const float* __restrict__ h_pre, const float* __restrict__ sum_h,
    const float* __restrict__ sumsq_h, const float* __restrict__ gamma_h,
    const float* __restrict__ beta_h, const float* __restrict__ W1,
    const float* __restrict__ b1, const float* __restrict__ W2,
    const float* __restrict__ b2, const float* __restrict__ h_in,
    float* __restrict__ h_out)
{
  __shared__ float hbn[16 * 132];   // pitch 132 (mod 64 == 4): no conflicts
  __shared__ float zbuf[16 * 516];  // pitch 516 (mod 64 == 4)
  const int tid = threadIdx.x, wave = tid >> 5, lane = tid & 31;
  const int half = lane >> 4, mr = lane & 15;
  const int m0 = blockIdx.x * 16;

  for (int idx = tid; idx < 16 * D; idx += 256) {
    const int r = idx >> 7, c = idx & 127;
    const float mean = sum_h[c] * (1.f / MH);
    const float var  = sumsq_h[c] * (1.f / MH) - mean * mean;
    const float sc   = gamma_h[c] * rsqrtf(var + EPSQ);
    hbn[r * 132 + c] = (h_pre[(m0 + r) * D + c] - mean) * sc + beta_h[c];
  }
  __syncthreads();

  const float c0 = 0.7978845608028654f;  // sqrt(2/pi)
  for (int t = 0; t < 4; ++t) {          // wave covers 64 of the 512 cols
    const int n0 = wave * 64 + t * 16;
    v8f acc = {};
    for (int k = 0; k < D; k += 4) {
      const int kk = k + 2 * half;
      v2f a = *(const v2f*)(&hbn[mr * 132 + kk]);
      v2f bf; bf.x = W1[kk * F + n0 + mr]; bf.y = W1[(kk + 1) * F + n0 + mr];
      acc = wmma_f32(a, bf, acc);
    }
    const float bb = b1[n0 + mr];
#pragma unroll
    for (int r = 0; r < 8; ++r) {
      const float x = acc[r] + bb;
      const float g = 0.5f * x * (1.f + tanhf(c0 * (x + 0.044715f * x * x * x)));
      zbuf[(r + 8 * half) * 516 + n0 + mr] = g;
    }
  }
  __syncthreads();

  {
    const int n0 = wave * 16;            // 8 waves cover 128 output cols
    v8f acc = {};
    for (int k = 0; k < F; k += 4) {
      const int kk = k + 2 * half;
      v2f a = *(const v2f*)(&zbuf[mr * 516 + kk]);
      v2f bf; bf.x = W2[kk * D + n0 + mr]; bf.y = W2[(kk + 1) * D + n0 + mr];
      acc = wmma_f32(a, bf, acc);
    }
    const float bb = b2[n0 + mr];
#pragma unroll
    for (int r = 0; r < 8; ++r) {
      const int idx = (m0 + r + 8 * half) * D + n0 + mr;
      h_out[idx] = h_in[idx] + acc[r] + bb;
    }
  }
}

// ----------------- k4: e_out = e_in + relu(BN(e_new)), streaming float4 pass
__global__ __launch_bounds__(256) void k4_efinal(
    const float* __restrict__ e_in, float* __restrict__ e_io,
    const float* __restrict__ sum_e, const float* __restrict__ sumsq_e,
    const float* __restrict__ gamma_e, const float* __restrict__ beta_e)
{
  const long n4 = (long)ME * D / 4;
  const long stride = (long)gridDim.x * blockDim.x;
  for (long idx = (long)blockIdx.x * blockDim.x + threadIdx.x; idx < n4;
       idx += stride) {
    const int c0 = (int)((idx * 4) & (D - 1));
    float4 v  = ((const float4*)e_io)[idx];
    float4 ei = ((const float4*)e_in)[idx];
    float vv[4] = {v.x, v.y, v.z, v.w};
    float ee[4] = {ei.x, ei.y, ei.z, ei.w};
    float o[4];
#pragma unroll
    for (int q = 0; q < 4; ++q) {
      const int c = c0 + q;
      const float mean = sum_e[c] * (1.f / (float)ME);
      const float var  = sumsq_e[c] * (1.f / (float)ME) - mean * mean;
      const float sc   = gamma_e[c] * rsqrtf(var + EPSQ);
      float y = (vv[q] - mean) * sc + beta_e[c];
      o[q] = ee[q] + fmaxf(y, 0.f);
    }
    ((float4*)e_io)[idx] = make_float4(o[0], o[1], o[2], o[3]);
  }
}

// ---------------------------------------------------------------------------
extern "C" void kernel_launch(void* const* d_in, const int* in_sizes, int n_in,
                              void* d_out, int out_size, void* d_ws,
                              size_t ws_size, hipStream_t stream) {
  const float* h  = (const float*)d_in[0];
  const float* e  = (const float*)d_in[1];
  const float* Uw = (const float*)d_in[2];  const float* Ub = (const float*)d_in[3];
  const float* Vw = (const float*)d_in[4];  const float* Vb = (const float*)d_in[5];
  const float* Aw = (const float*)d_in[6];  const float* Ab = (const float*)d_in[7];
  const float* Bw = (const float*)d_in[8];  const float* Bb = (const float*)d_in[9];
  const float* Cw = (const float*)d_in[10]; const float* Cb = (const float*)d_in[11];
  const float* gamma_h = (const float*)d_in[12];
  const float* beta_h  = (const float*)d_in[13];
  const float* gamma_e = (const float*)d_in[14];
  const float* beta_e  = (const float*)d_in[15];
  const float* W1 = (const float*)d_in[16]; const float* b1 = (const float*)d_in[17];
  const float* W2 = (const float*)d_in[18]; const float* b2 = (const float*)d_in[19];

  float* ws      = (float*)d_ws;          // ~6.3 MB used
  float* Uh      = ws;                    // [2048,128] x4 contiguous
  float* h_agg   = ws + 4 * MH * D;       // [2048,128]
  float* h_pre   = h_agg + MH * D;        // [2048,128]
  float* sum_e   = h_pre + MH * D;        // [128]
  float* sumsq_e = sum_e + 128;
  float* sum_h   = sum_e + 256;
  float* sumsq_h = sum_e + 384;
  float* Vh = Uh + (size_t)MH * D;
  float* Ah = Uh + (size_t)2 * MH * D;
  float* Bh = Uh + (size_t)3 * MH * D;

  float* h_out = (float*)d_out;
  float* e_tmp = (float*)d_out + MH * D;  // e region, used as e_new temp

  const int nzero = 2 * MH * D + 512;     // h_agg + h_pre(+stats), one shot
  k0_zero<<<(nzero + 255) / 256, 256, 0, stream>>>(h_agg, nzero);
  k1_proj<<<512, 256, 0, stream>>>(h, Uw, Ub, Vw, Vb, Aw, Ab, Bw, Bb, Uh);
  k2_edge<<<2048, 512, 0, stream>>>(e, Cw, Cb, Ah, Bh, Vh, e_tmp, h_agg,
                                    sum_e, sumsq_e);
  k3a_hstats<<<64, 256, 0, stream>>>(Uh, h_agg, h_pre, sum_h, sumsq_h);
  k3b_ffn<<<128, 256, 0, stream>>>(h_pre, sum_h, sumsq_h, gamma_h, beta_h, W1,
                                   b1, W2, b2, h, h_out);
  k4_efinal<<<8192, 256, 0, stream>>>(e, e_tmp, sum_e, sumsq_e, gamma_e,
                                      beta_e);
}